// _Sobel3D_74139725463914
// MI455X (gfx1250) — compile-verified
//
#include <hip/hip_runtime.h>
#include <hip/hip_bf16.h>

// ---------------------------------------------------------------------------
// 3D Sobel (depthwise 3x3x3, zero pad) on (N=2, C=16, D=96, H=128, W=128) f32.
// Outputs (gx, gy, gz) concatenated flat in d_out.
//
// Memory-bound: ~805 MB min traffic -> ~35 us @ 23.3 TB/s; flops after
// separable factorization ~1.3 GFLOP (negligible). WMMA would add data
// movement; the CDNA5 feature used instead is the async global->LDS DMA
// (global_load_async_to_lds_b32) with ASYNCcnt-based software pipelining
// (s_wait_asynccnt 2 keeps two halo planes in flight) while the separable
// reduction runs out of a 3-plane register ring.
// ---------------------------------------------------------------------------

namespace {
constexpr int Dz = 96;
constexpr int Hy = 128;
constexpr int Wx = 128;
constexpr int NSLAB = 2 * 16;               // N * C independent channels
constexpr int TH = 8;                       // tile height (y)
constexpr int TW = 32;                      // tile width  (x)
constexpr int NTHREADS = TH * TW;           // 256 threads = 8 wave32
constexpr int HALO_H = TH + 2;              // 10
constexpr int HALO_W = TW + 2;              // 34
constexpr int PLANE = HALO_H * HALO_W;      // 340 floats per halo plane
constexpr int HWp = Hy * Wx;                // 16384
}  // namespace

__device__ __forceinline__ int iclamp(int v, int lo, int hi) {
  return v < lo ? lo : (v > hi ? hi : v);
}

// Raw LDS byte offset of a __shared__ object (ptrtoint of the AS(3) pointer).
__device__ __forceinline__ unsigned lds_byte_off(const void* p) {
  return (unsigned)(unsigned long long)(__attribute__((address_space(3))) const void*)p;
}

// Async DMA: copies 4 bytes at global sbase + voff to LDS byte address lds_addr   (GVS addressing, ASYNCcnt)
__device__ __forceinline__ void async_load_b32(unsigned lds_addr, unsigned voff,
                                               const float* sbase) {
  asm volatile("global_load_async_to_lds_b32 %0, %1, %2"
               :: "v"(lds_addr), "v"(voff), "s"(sbase)
               : "memory");
}

__device__ __forceinline__ void wait_async_le2() {
  asm volatile("s_wait_asynccnt 2" ::: "memory");
}

__global__ __launch_bounds__(NTHREADS) void _Sobel3D_async_kernel(
    const float* __restrict__ x, float* __restrict__ out) {
  __shared__ float lds[3][PLANE];           // triple-buffered halo planes, 4080 B

  const int tid  = threadIdx.x;
  const int tw0  = blockIdx.x * TW;
  const int th0  = blockIdx.y * TH;
  const int slab = blockIdx.z;              // 0..31, uniform -> SGPR math

  const size_t VOL  = (size_t)Dz * HWp;     // per-slab voxels
  const size_t OUTN = (size_t)NSLAB * VOL;  // one output tensor
  const float* base = x + (size_t)slab * VOL;

  // ---- DMA loader slots: 340 halo cells over 256 lanes (2 per lane) --------
  const int s0 = tid;
  const int s1 = (tid + NTHREADS < PLANE) ? (tid + NTHREADS) : (PLANE - 1);
  // plane-invariant, address-clamped global (h,w) offset per slot
  const int off0 = iclamp(th0 + s0 / HALO_W - 1, 0, Hy - 1) * Wx +
                   iclamp(tw0 + s0 % HALO_W - 1, 0, Wx - 1);
  const int off1 = iclamp(th0 + s1 / HALO_W - 1, 0, Hy - 1) * Wx +
                   iclamp(tw0 + s1 % HALO_W - 1, 0, Wx - 1);
  const unsigned lbase = lds_byte_off(&lds[0][0]);
  const unsigned la0   = lbase + (unsigned)s0 * 4u;
  const unsigned la1   = lbase + (unsigned)s1 * 4u;

  // ---- compute-pixel coords + boundary masks (zero padding) ----------------
  const int lx = tid % TW, ly = tid / TW;
  const int gw = tw0 + lx, gh = th0 + ly;
  const float mL = (gw > 0)      ? 1.0f : 0.0f;
  const float mR = (gw < Wx - 1) ? 1.0f : 0.0f;
  const float mU = (gh > 0)      ? 1.0f : 0.0f;
  const float mD = (gh < Hy - 1) ? 1.0f : 0.0f;
  const size_t outPix = (size_t)slab * VOL + (size_t)gh * Wx + gw;

  float* __restrict__ gx = out;
  float* __restrict__ gy = out + OUTN;
  float* __restrict__ gz = out + 2 * OUTN;

  // issue DMA of plane p into LDS buffer `slot` (z address-clamped; planes
  // past the end are masked to zero in compute, never in the DMA)
  auto issue = [&](int p, int slot) {
    const int zc = (p < Dz) ? p : (Dz - 1);
    const unsigned zb = (unsigned)(zc * HWp) * 4u;
    const unsigned sb = (unsigned)(slot * PLANE) * 4u;
    async_load_b32(la0 + sb, zb + (unsigned)off0 * 4u, base);
    async_load_b32(la1 + sb, zb + (unsigned)off1 * 4u, base);
  };

  // register ring of in-plane partials: A = sy*sx, B = sy*dx, C = dy*sx
  float A[3], B[3], C[3];
  auto computeABC = [&](int p, int slot) {
    const float mz = (p < Dz) ? 1.0f : 0.0f;       // uniform z zero-pad mask
    const float* pb = &lds[slot][0];
    float sx[3], dx[3];
#pragma unroll
    for (int r = 0; r < 3; ++r) {                  // rows gh-1, gh, gh+1
      float l = pb[(ly + r) * HALO_W + lx];
      float c = pb[(ly + r) * HALO_W + lx + 1];
      float q = pb[(ly + r) * HALO_W + lx + 2];
      l *= mL;
      q *= mR;
      sx[r] = l + 2.0f * c + q;                    // smooth_x
      dx[r] = q - l;                               // deriv_x
    }
    sx[0] *= mU; dx[0] *= mU;
    sx[2] *= mD; dx[2] *= mD;
    A[slot] = (sx[0] + 2.0f * sx[1] + sx[2]) * mz; // smooth_y(smooth_x)
    B[slot] = (dx[0] + 2.0f * dx[1] + dx[2]) * mz; // smooth_y(deriv_x)
    C[slot] = (sx[2] - sx[0]) * mz;                // deriv_y(smooth_x)
  };

  // one z-step: needs plane z+1 resident; keeps planes z+2, z+3 in flight
  auto step = [&](int z, int im1, int i0, int ip1) {
    wait_async_le2();                // plane z+1 DMA complete (this wave)
    __syncthreads();                 // ... across all 8 waves; also fences the
                                     // last reader of buffer i0 (plane z)
    issue(z + 3, i0);                // refill freed buffer while we compute
    computeABC(z + 1, ip1);
    const size_t idx = outPix + (size_t)z * HWp;
    gx[idx] = B[im1] + 2.0f * B[i0] + B[ip1];      // smooth_z(B)
    gy[idx] = C[im1] + 2.0f * C[i0] + C[ip1];      // smooth_z(C)
    gz[idx] = A[ip1] - A[im1];                     // deriv_z(A)
  };

  // ---- prologue: planes 0 and 1 in flight, compute plane 0, ring[-1]=0 -----
  issue(0, 0);
  issue(1, 1);
  wait_async_le2();                  // plane 0 done (in-order completion)
  __syncthreads();
  computeABC(0, 0);
  A[2] = 0.0f; B[2] = 0.0f; C[2] = 0.0f;           // plane z = -1 (zero pad)
  issue(2, 2);

  // ---- z sweep, ring indices rotated by hand so A/B/C stay in VGPRs --------
  for (int z = 0; z < Dz; z += 3) {
    step(z + 0, 2, 0, 1);
    step(z + 1, 0, 1, 2);
    step(z + 2, 1, 2, 0);
  }
}

extern "C" void kernel_launch(void* const* d_in, const int* in_sizes, int n_in,
                              void* d_out, int out_size, void* d_ws, size_t ws_size,
                              hipStream_t stream) {
  (void)in_sizes; (void)n_in; (void)d_ws; (void)ws_size; (void)out_size;
  const float* x = (const float*)d_in[0];
  float* out = (float*)d_out;
  dim3 grid(Wx / TW, Hy / TH, NSLAB);              // 4 x 16 x 32 = 2048 blocks
  _Sobel3D_async_kernel<<<grid, NTHREADS, 0, stream>>>(x, out);
}